// CPN_mc_68530498175131
// MI455X (gfx1250) — compile-verified
//
#include <hip/hip_runtime.h>
#include <stdint.h>

// MTLU: out = w[c][bin]*x + b[c][bin], bin = clamp(floor(x/0.1)+10, 0, 19)
// Memory-bound: 536 MB @ 23.3 TB/s => ~23 us floor. No matmul structure -> no WMMA.
// CDNA5 paths used: global_load_async_to_lds_b32 (ASYNCcnt) table staging,
// ds_load_b64 float2 gathers, NT b128 streaming loads/stores, global_prefetch_b8.

typedef float __attribute__((ext_vector_type(4))) f4;

#define BIN_NUM  20
#define HALF     10
#define FEAT     64
#define TABLE    (FEAT * BIN_NUM)   // 1280 entries
#define NTHREADS 256
#define NBLOCKS  4096

__device__ __forceinline__ float mtlu_one(float xv, const float2* __restrict__ tbl, int base) {
    // bin = clamp(floor(x / 0.1f) + 10, 0, 19)  (precise division to match reference)
    int idx = (int)floorf(xv / 0.1f) + HALF;
    idx = min(max(idx, 0), BIN_NUM - 1);
    float2 wb = tbl[base + idx];          // ds_load_b64 gather: {w, b}
    return fmaf(wb.x, xv, wb.y);
}

__global__ __launch_bounds__(NTHREADS) void mtlu_kernel(
    const float* __restrict__ x,
    const float* __restrict__ mtlu_y,
    const float* __restrict__ mtlu_y_,
    float* __restrict__ out,
    int n4)
{
    __shared__ float  s_y [TABLE];
    __shared__ float  s_y2[TABLE];
    __shared__ float2 s_wb[TABLE];   // fused {w, b}, 8B-aligned for ds_load_b64

    const int t = threadIdx.x;

    // ---- Stage the two 1280-float tables into LDS via gfx1250 async loads ----
    // (5 b32 async loads per thread per table; tracked by ASYNCcnt)
    for (int e = t; e < TABLE; e += NTHREADS) {
        uint64_t ga = (uint64_t)(uintptr_t)(mtlu_y  + e);
        uint32_t la = (uint32_t)(uintptr_t)(&s_y[e]);     // low 32 bits = LDS offset
        asm volatile("global_load_async_to_lds_b32 %0, %1, off"
                     :: "v"(la), "v"(ga) : "memory");
        uint64_t gb = (uint64_t)(uintptr_t)(mtlu_y_ + e);
        uint32_t lb = (uint32_t)(uintptr_t)(&s_y2[e]);
        asm volatile("global_load_async_to_lds_b32 %0, %1, off"
                     :: "v"(lb), "v"(gb) : "memory");
    }
    asm volatile("s_wait_asynccnt 0" ::: "memory");
    __syncthreads();

    // ---- Fuse per-bin slope/intercept: w=(y-y_)/0.1 ; b=y-(y-y_)*(bin-9) ----
    for (int e = t; e < TABLE; e += NTHREADS) {
        float yy = s_y[e];
        float yo = s_y2[e];
        int   bin = e % BIN_NUM;
        float d  = yy - yo;
        float w  = d / 0.1f;
        float b  = fmaf(-d, (float)(bin - (HALF - 1)), yy);
        s_wb[e] = make_float2(w, b);
    }
    __syncthreads();

    // ---- Streaming main loop: 128-bit NT load -> 4x (LUT + FMA) -> NT store ----
    const f4* __restrict__ xin  = (const f4*)x;
    f4*       __restrict__ xout = (f4*)out;
    const int stride = (int)(gridDim.x * blockDim.x);

    for (int i = (int)(blockIdx.x * blockDim.x) + t; i < n4; i += stride) {
        if (i + stride < n4)
            __builtin_prefetch((const void*)(xin + i + stride), 0, 0); // global_prefetch_b8

        f4 v = __builtin_nontemporal_load(xin + i);      // global_load_b128 th:NT

        // 16384 float4s per (batch,channel) plane -> channel uniform per vector
        const int base = ((i >> 14) & (FEAT - 1)) * BIN_NUM;

        f4 r;
        r.x = mtlu_one(v.x, s_wb, base);
        r.y = mtlu_one(v.y, s_wb, base);
        r.z = mtlu_one(v.z, s_wb, base);
        r.w = mtlu_one(v.w, s_wb, base);

        __builtin_nontemporal_store(r, xout + i);        // global_store_b128 th:NT
    }
}

extern "C" void kernel_launch(void* const* d_in, const int* in_sizes, int n_in,
                              void* d_out, int out_size, void* d_ws, size_t ws_size,
                              hipStream_t stream) {
    const float* x       = (const float*)d_in[0];  // (16, 64, 256, 256) f32
    const float* mtlu_y  = (const float*)d_in[1];  // (64, 20) f32
    const float* mtlu_y_ = (const float*)d_in[2];  // (64, 20) f32
    float*       out     = (float*)d_out;          // same shape as x, f32

    const int n4 = out_size / 4;                   // 16,777,216 float4s (exact)
    mtlu_kernel<<<dim3(NBLOCKS), dim3(NTHREADS), 0, stream>>>(x, mtlu_y, mtlu_y_, out, n4);
}